// Attention_63170378989903
// MI455X (gfx1250) — compile-verified
//
#include <hip/hip_runtime.h>
#include <math.h>

// MI455X (gfx1250) paged-attention decode, fp32, wave32 + V_WMMA_F32_16X16X4_F32.
//
// Bandwidth-bound (~1 GB KV stream, AI ~= 2 flop/B -> ~45us floor @ 23.3 TB/s).
// 512 workgroups (one per (b, kv_head)) x 8 wave32s; each wave flash-attends a
// strided subset of 16-token pages, partials merged via LDS.
// The new-token k/v row is substituted at read time (slot == ctx-1 of its own
// sequence), so the paged caches are never written (inputs stay immutable).
//
// Round-2 changes: Q^T fragments live in LDS (shared by all waves) instead of
// 64 VGPRs; the freed registers stage the whole K page in kf[32] so the QK
// stream runs as one 32-deep load clause instead of 32 serialized loads.

typedef __attribute__((ext_vector_type(2))) float v2f;
typedef __attribute__((ext_vector_type(8))) float v8f;

#define B_    64
#define H_    32
#define HK_   8
#define G_    4
#define D_    128
#define BS_   16
#define BPS_  128              // blocks (pages) per sequence
#define NW_   8                // waves per workgroup
#define SCALE_ 0.08838834764831845f

__launch_bounds__(256, 1)
__global__ void paged_attn_decode_kernel(
    const float* __restrict__ q,            // [B, H, D]
    const float* __restrict__ knew,         // [B, HK, D]
    const float* __restrict__ vnew,         // [B, HK, D]
    const float* __restrict__ kc,           // [NB, BS, HK, D]
    const float* __restrict__ vc,           // [NB, BS, HK, D]
    const int*   __restrict__ block_tables, // [B, BPS]
    const int*   __restrict__ context_lens, // [B]
    float* __restrict__ out)                // [B, H*D]
{
  __shared__ v2f   q_lds[32][32];            // Q^T B-fragments: [t-step][lane]
  __shared__ float p_buf[NW_][BS_][17];      // per-wave P transpose tile (+pad)
  __shared__ float red_m[NW_][G_];
  __shared__ float red_l[NW_][G_];
  __shared__ float red_o[NW_][G_][D_];

  const int bh   = blockIdx.x;
  const int b    = bh >> 3;
  const int hk   = bh & 7;
  const int tid  = threadIdx.x;
  const int wave = tid >> 5;
  const int lane = tid & 31;
  const int g16  = lane & 15;     // N-column (g for QK-C, d%16 for PV-B)
  const int hi   = lane >> 4;     // which half-wave
  const int hi2  = hi * 2;        // K-offset inside a 4-wide WMMA step
  const int hi8  = hi * 8;        // M-offset inside C rows

  const int ctx    = context_lens[b];
  const int last   = ctx - 1;     // global position holding the new token
  const int npages = (ctx + BS_ - 1) / BS_;

  const float* kcur   = knew + (size_t)(b * HK_ + hk) * D_;
  const float* vcur   = vnew + (size_t)(b * HK_ + hk) * D_;
  const int*   bt_row = block_tables + (size_t)b * BPS_;

  // ---- Build Q^T B-fragments in LDS once (shared across all 8 waves).
  // B 4x16 layout: lane l holds B[K=hi2+r][N=l&15] = Q[g=l&15][t*4+hi2+r].
  {
    const bool   gv = (g16 < G_);
    const float* qb = q + (size_t)(b * H_ + hk * G_ + (gv ? g16 : 0)) * D_ + hi2;
#pragma unroll
    for (int i = 0; i < 4; ++i) {
      const int t = wave * 4 + i;           // 8 waves x 4 = all 32 t-steps
      v2f vq = *(const v2f*)(qb + t * 4);
      if (!gv) { vq.x = 0.0f; vq.y = 0.0f; }
      q_lds[t][lane] = vq;
    }
  }
  __syncthreads();

  // Flash-attention running state (per lane: one g-column, duplicated halves).
  v8f acc[8];                    // PV accumulators: C2[g][d16] per 16-d chunk
#pragma unroll
  for (int k = 0; k < 8; ++k) acc[k] = (v8f){};
  float m_run = -INFINITY;
  float l_run = 0.0f;

  // Software-pipelined block-id fetch: blk for page p is loaded one iter early.
  int blk_cur = (wave < npages) ? bt_row[wave] : 0;

  for (int p = wave; p < npages; p += NW_) {
    // Stop the compiler from hoisting per-page LDS reads into live VGPRs.
    asm volatile("" ::: "memory");

    const int blk = blk_cur;
    const bool have_next = (p + NW_ < npages);
    if (have_next) blk_cur = bt_row[p + NW_];     // consumed next iteration

    // Per-lane K row pointer (A-matrix row M = g16), new-token substitution.
    const int    sK   = p * BS_ + g16;
    const float* krow = (sK == last)
        ? kcur
        : (kc + (((size_t)blk * BS_ + g16) * HK_ + hk) * D_);

    // ---- Stage the whole K page: 32 independent b64 loads (one clause) ----
    v2f kf[32];
#pragma unroll
    for (int t = 0; t < 32; ++t)
      kf[t] = *(const v2f*)(krow + t * 4 + hi2);

    // Prefetch next page's K row (next blk id has landed by now anyway).
    if (have_next)
      __builtin_prefetch(kc + (((size_t)blk_cur * BS_ + g16) * HK_ + hk) * D_, 0, 1);

    // ---- QK^T: C1[s][g] = K(16x4) x Q^T(4x16); drain kf with WMMAs ----
    v8f sc = (v8f){};
#pragma unroll
    for (int grp = 0; grp < 4; ++grp) {
      v2f qg[8];
#pragma unroll
      for (int i = 0; i < 8; ++i) qg[i] = q_lds[grp * 8 + i][lane];
#pragma unroll
      for (int i = 0; i < 8; ++i)
        sc = __builtin_amdgcn_wmma_f32_16x16x4_f32(
            false, kf[grp * 8 + i], false, qg[i], (short)0, sc, false, false);
    }

    // ---- scale + length mask + online softmax (per g-column) ----
    float pe[8];
    float mu = -INFINITY;
#pragma unroll
    for (int r = 0; r < 8; ++r) {
      const int sg = p * BS_ + r + hi8;
      const float v = (sg < ctx) ? sc[r] * SCALE_ : -INFINITY;
      pe[r] = v;
      mu = fmaxf(mu, v);
    }
    mu = fmaxf(mu, __shfl_xor(mu, 16, 32));        // combine both half-waves
    const float m_new = fmaxf(m_run, mu);
    const float corr  = __expf(m_run - m_new);
    float rsum = 0.0f;
#pragma unroll
    for (int r = 0; r < 8; ++r) {
      const float e = __expf(pe[r] - m_new);
      pe[r] = e;
      rsum += e;
    }
    rsum += __shfl_xor(rsum, 16, 32);
    l_run = l_run * corr + rsum;
    m_run = m_new;

    // Broadcast per-row corrections (rows g=0..3 carry data; rest stay 0).
    const float c0 = __shfl(corr, 0, 32);
    const float c1 = __shfl(corr, 1, 32);
    const float c2 = __shfl(corr, 2, 32);
    const float c3 = __shfl(corr, 3, 32);
#pragma unroll
    for (int k = 0; k < 8; ++k) {
      acc[k][0] *= c0; acc[k][1] *= c1; acc[k][2] *= c2; acc[k][3] *= c3;
    }

    // ---- transpose P via per-wave LDS tile (zero dummy rows g>=4) ----
#pragma unroll
    for (int r = 0; r < 8; ++r)
      p_buf[wave][r + hi8][g16] = (g16 < G_) ? pe[r] : 0.0f;

    // ---- PV: C2[g][d16] += P^T(16x4) x V(4x16), 4 s-steps x 8 d-chunks ----
#pragma unroll
    for (int s0 = 0; s0 < BS_; s0 += 4) {
      v2f pa;                                      // A: P^T[g=g16][s0+hi2 .. +1]
      pa.x = p_buf[wave][s0 + hi2][g16];
      pa.y = p_buf[wave][s0 + hi2 + 1][g16];

      const int    r0    = p * BS_ + s0 + hi2;
      const float* vrow0 = (r0 == last)
          ? vcur : (vc + (((size_t)blk * BS_ + s0 + hi2) * HK_ + hk) * D_);
      const float* vrow1 = (r0 + 1 == last)
          ? vcur : (vc + (((size_t)blk * BS_ + s0 + hi2 + 1) * HK_ + hk) * D_);

      // Stage all 16 V values first (keeps 16 loads in flight), then 8 WMMAs.
      v2f vf[8];
#pragma unroll
      for (int k = 0; k < 8; ++k) {
        vf[k].x = vrow0[k * 16 + g16];             // B: V[s0+hi2+r][k*16+g16]
        vf[k].y = vrow1[k * 16 + g16];
      }
#pragma unroll
      for (int k = 0; k < 8; ++k)
        acc[k] = __builtin_amdgcn_wmma_f32_16x16x4_f32(
            false, pa, false, vf[k], (short)0, acc[k], false, false);
    }
  }

  // ---- publish per-wave partials ----
  if (lane < G_) {                       // lanes 0..3 hold g=0..3 state
    red_m[wave][lane] = m_run;
    red_l[wave][lane] = l_run;
  }
  if (lane < 16) {                       // C2 rows 0..3 live in low half, vgpr r
#pragma unroll
    for (int k = 0; k < 8; ++k) {
      red_o[wave][0][k * 16 + lane] = acc[k][0];
      red_o[wave][1][k * 16 + lane] = acc[k][1];
      red_o[wave][2][k * 16 + lane] = acc[k][2];
      red_o[wave][3][k * 16 + lane] = acc[k][3];
    }
  }
  __syncthreads();

  // ---- cross-wave softmax merge + write ----
  for (int o = tid; o < G_ * D_; o += 256) {
    const int g = o >> 7;
    const int d = o & (D_ - 1);
    float M = -INFINITY;
#pragma unroll
    for (int w = 0; w < NW_; ++w) M = fmaxf(M, red_m[w][g]);
    float L = 0.0f, O = 0.0f;
#pragma unroll
    for (int w = 0; w < NW_; ++w) {
      const float e = __expf(red_m[w][g] - M);
      L += e * red_l[w][g];
      O += e * red_o[w][g][d];
    }
    out[(size_t)b * (H_ * D_) + (hk * G_ + g) * D_ + d] = O / L;
  }
}

extern "C" void kernel_launch(void* const* d_in, const int* in_sizes, int n_in,
                              void* d_out, int out_size, void* d_ws, size_t ws_size,
                              hipStream_t stream) {
  (void)in_sizes; (void)n_in; (void)d_ws; (void)ws_size; (void)out_size;
  const float* q    = (const float*)d_in[0];
  const float* k    = (const float*)d_in[1];
  const float* v    = (const float*)d_in[2];
  const float* kc   = (const float*)d_in[3];
  const float* vc   = (const float*)d_in[4];
  // d_in[5] = slot_mapping: unused — the new token is read-time substituted
  const int* bt     = (const int*)d_in[6];
  const int* ctxl   = (const int*)d_in[7];
  float* out        = (float*)d_out;

  paged_attn_decode_kernel<<<dim3(B_ * HK_), dim3(256), 0, stream>>>(
      q, k, v, kc, vc, bt, ctxl, out);
}